// MNISTNET_51470888075432
// MI455X (gfx1250) — compile-verified
//
#include <hip/hip_runtime.h>

// ---------------------------------------------------------------------------
// LSH sampled layer for MI455X (gfx1250, wave32, WMMA)
//   H   = relu(X @ W1[:, ids] + b1[ids])        [512 x 8192]
//   out = H @ W2[:, ids]^T + b2                 [512 x 10]
// fp32 WMMA (16x16x4): the op is bound by the random column gather out of
// the 822MB W1 (~400MB effective HBM traffic at one line per 4B element), so
// the fp32 matrix path keeps reference precision at zero throughput cost.
// Wave tile 32x64 (2x4 WMMA tiles) -> 12 loads : 8 WMMAs in the K-loop.
// ---------------------------------------------------------------------------

typedef float v2f __attribute__((ext_vector_type(2)));
typedef float v8f __attribute__((ext_vector_type(8)));

#define NROWS  512
#define DDIM   784
#define SSAMP  8192
#define CCLS   10
#define HID    262144

// --- Xt[d][n] = X[n][d] : put X K-major so WMMA A-frag loads are coalesced --
__global__ __launch_bounds__(256) void k_transpose_x(
    const float* __restrict__ X, float* __restrict__ Xt) {
  int idx = blockIdx.x * blockDim.x + threadIdx.x;   // over N*D (exact)
  if (idx >= NROWS * DDIM) return;
  int n = idx / DDIM;
  int d = idx - n * DDIM;
  Xt[d * NROWS + n] = X[idx];
}

// --- WgT[d][s] = W1[d][ids[s]] : K-major gathered weights, coalesced writes -
__global__ __launch_bounds__(256) void k_gather_w1(
    const float* __restrict__ W1, const int* __restrict__ ids,
    float* __restrict__ WgT) {
  int idx = blockIdx.x * blockDim.x + threadIdx.x;   // over D*S (exact)
  int d = idx >> 13;               // /8192
  int s = idx & (SSAMP - 1);
  WgT[idx] = W1[(size_t)d * HID + (unsigned)ids[s]];
}

// --- bg[s] = b1[ids[s]];  W2g[s][c] = W2[c][ids[s]] ------------------------
__global__ __launch_bounds__(256) void k_gather_small(
    const float* __restrict__ b1, const float* __restrict__ W2,
    const int* __restrict__ ids, float* __restrict__ bg,
    float* __restrict__ W2g) {
  int s = blockIdx.x * blockDim.x + threadIdx.x;
  if (s >= SSAMP) return;
  unsigned id = (unsigned)ids[s];
  bg[s] = b1[id];
#pragma unroll
  for (int c = 0; c < CCLS; ++c)
    W2g[s * CCLS + c] = W2[(size_t)c * HID + id];
}

// --- GEMM1: H[n][s] = relu(sum_k Xt[k][n] * WgT[k][s] + bg[s]) -------------
// Wave tile 32x64 = 2x4 v_wmma_f32_16x16x4_f32; 8 waves/block -> 64x256 tile.
// A-frag (16x4 f32): lanes 0-15 M=0..15; VGPR0=K{0|2}, VGPR1=K{1|3} by half.
// B-frag (4x16 f32): mirrored; all frag loads are 16-lane contiguous dwords.
__global__ __launch_bounds__(256) void k_gemm1_wmma(
    const float* __restrict__ Xt,    // [D][N]
    const float* __restrict__ WgT,   // [D][S]
    const float* __restrict__ bg,    // [S]
    float* __restrict__ H) {         // [N][S]
  const int tid  = threadIdx.x;
  const int wave = tid >> 5;
  const int lane = tid & 31;
  const int half = lane >> 4;        // 0: lanes 0-15, 1: lanes 16-31
  const int lm   = lane & 15;

  const int m0 = blockIdx.y * 64  + (wave & 1)  * 32;   // row tile in X
  const int n0 = blockIdx.x * 256 + (wave >> 1) * 64;   // col tile in S

  v8f acc[2][4];
#pragma unroll
  for (int i = 0; i < 2; ++i)
#pragma unroll
    for (int j = 0; j < 4; ++j) acc[i][j] = (v8f){};

#pragma unroll 2
  for (int k = 0; k < DDIM; k += 4) {
    const int kr = k + 2 * half;                 // this half-wave's K rows
    v2f a[2], b[4];
#pragma unroll
    for (int i = 0; i < 2; ++i) {
      a[i].x = Xt[kr       * NROWS + m0 + 16 * i + lm];
      a[i].y = Xt[(kr + 1) * NROWS + m0 + 16 * i + lm];
    }
#pragma unroll
    for (int j = 0; j < 4; ++j) {
      b[j].x = WgT[kr       * SSAMP + n0 + 16 * j + lm];
      b[j].y = WgT[(kr + 1) * SSAMP + n0 + 16 * j + lm];
    }
#pragma unroll
    for (int i = 0; i < 2; ++i)
#pragma unroll
      for (int j = 0; j < 4; ++j)
        acc[i][j] = __builtin_amdgcn_wmma_f32_16x16x4_f32(
            false, a[i], false, b[j], (short)0, acc[i][j], false, false);
  }

  // C/D layout: VGPR r -> M = r (lanes 0-15) or M = 8+r (lanes 16-31), N = lm
  float bias[4];
  int   col[4];
#pragma unroll
  for (int j = 0; j < 4; ++j) {
    col[j]  = n0 + 16 * j + lm;
    bias[j] = bg[col[j]];
  }
#pragma unroll
  for (int i = 0; i < 2; ++i) {
#pragma unroll
    for (int r = 0; r < 8; ++r) {
      const int row = m0 + 16 * i + r + half * 8;
#pragma unroll
      for (int j = 0; j < 4; ++j) {
        float v = acc[i][j][r] + bias[j];
        H[row * SSAMP + col[j]] = v > 0.f ? v : 0.f;
      }
    }
  }
}

// --- GEMM2: out[n][c] = sum_s H[n][s] * W2g[s][c] + b2[c] (block per row) --
__global__ __launch_bounds__(256) void k_out(
    const float* __restrict__ H, const float* __restrict__ W2g,
    const float* __restrict__ b2, float* __restrict__ out) {
  __shared__ float red[CCLS][257];
  const int n = blockIdx.x;
  const int tid = threadIdx.x;
  float acc[CCLS];
#pragma unroll
  for (int c = 0; c < CCLS; ++c) acc[c] = 0.f;

  for (int s = tid; s < SSAMP; s += 256) {
    const float h = H[n * SSAMP + s];
    const float* w = W2g + s * CCLS;
#pragma unroll
    for (int c = 0; c < CCLS; ++c) acc[c] = fmaf(h, w[c], acc[c]);
  }
#pragma unroll
  for (int c = 0; c < CCLS; ++c) red[c][tid] = acc[c];
  __syncthreads();
  for (int off = 128; off > 0; off >>= 1) {
    if (tid < off) {
#pragma unroll
      for (int c = 0; c < CCLS; ++c) red[c][tid] += red[c][tid + off];
    }
    __syncthreads();
  }
  if (tid < CCLS) out[n * CCLS + tid] = red[tid][0] + b2[tid];
}

// ---------------------------------------------------------------------------
extern "C" void kernel_launch(void* const* d_in, const int* in_sizes, int n_in,
                              void* d_out, int out_size, void* d_ws, size_t ws_size,
                              hipStream_t stream) {
  const float* X   = (const float*)d_in[0];
  const int*   ids = (const int*)  d_in[1];
  const float* W1  = (const float*)d_in[2];
  const float* b1  = (const float*)d_in[3];
  const float* W2  = (const float*)d_in[4];
  const float* b2  = (const float*)d_in[5];
  float* out = (float*)d_out;

  // workspace carve-up (floats): WgT | H | Xt | bg | W2g  (~44.4 MB total)
  float* ws  = (float*)d_ws;
  float* WgT = ws;                                      // D*S
  float* H   = WgT + (size_t)DDIM * SSAMP;              // N*S
  float* Xt  = H   + (size_t)NROWS * SSAMP;             // D*N
  float* bg  = Xt  + (size_t)DDIM * NROWS;              // S
  float* W2g = bg  + SSAMP;                             // S*C

  k_transpose_x <<<(NROWS * DDIM + 255) / 256, 256, 0, stream>>>(X, Xt);
  k_gather_w1   <<<(DDIM * SSAMP) / 256,       256, 0, stream>>>(W1, ids, WgT);
  k_gather_small<<<SSAMP / 256,                256, 0, stream>>>(b1, W2, ids, bg, W2g);
  k_gemm1_wmma  <<<dim3(SSAMP / 256, NROWS / 64), 256, 0, stream>>>(Xt, WgT, bg, H);
  k_out         <<<NROWS,                      256, 0, stream>>>(H, W2g, b2, out);
}